// TopK_33895881900714
// MI455X (gfx1250) — compile-verified
//
#include <hip/hip_runtime.h>

#define ROWLEN 24576
#define TPB    1024
#define KSEL   64
#define CHUNKS ((ROWLEN / 4) / TPB)   // 6 float4 chunks per thread

typedef float v4f __attribute__((ext_vector_type(4)));
typedef int   b128_t __attribute__((vector_size(16)));   // matches builtin param type
typedef __attribute__((address_space(1))) b128_t* gptr128_t;
typedef __attribute__((address_space(3))) b128_t* lptr128_t;

#if __has_builtin(__builtin_amdgcn_global_load_async_to_lds_b128)
#define HAVE_ASYNC_LDS 1
#endif

// Monotonic mapping: u32 key order == float order (totally ordered, no NaNs in input)
__device__ __forceinline__ unsigned fkey(float f) {
    unsigned u = __float_as_uint(f);
    unsigned m = (unsigned)(((int)u) >> 31) | 0x80000000u;
    return u ^ m;
}

__device__ __forceinline__ unsigned ballot32(bool p) {
#if __has_builtin(__builtin_amdgcn_ballot_w32)
    return __builtin_amdgcn_ballot_w32(p);
#else
    return (unsigned)__ballot(p);
#endif
}

__global__ __launch_bounds__(TPB, 1) void topk_scatter_kernel(
        const float* __restrict__ x, float* __restrict__ out) {
    __shared__ float    row[ROWLEN];          // 96 KB: full row staged in LDS
    __shared__ unsigned flags[ROWLEN / 32];   // 3 KB: tie-break selection bitset
    __shared__ unsigned hist[256];            // 1 KB: radix histogram
    __shared__ unsigned s_sel, s_want;

    const int tid = threadIdx.x;
    const size_t rbase = (size_t)blockIdx.x * ROWLEN;
    const float* __restrict__ xr = x + rbase;
    float* __restrict__ orow = out + rbase;

    // ---- zero the tie-break bitset -------------------------------------
    for (int i = tid; i < ROWLEN / 32; i += TPB) flags[i] = 0u;

    // ---- stage row into LDS via async global->LDS DMA path --------------
#ifdef HAVE_ASYNC_LDS
    for (int i = 0; i < CHUNKS; ++i) {
        const int chunk = i * TPB + tid;
        float* gsrc = const_cast<float*>(xr) + (size_t)chunk * 4;
        __builtin_amdgcn_global_load_async_to_lds_b128(
            (gptr128_t)gsrc,
            (lptr128_t)(&row[chunk * 4]),
            0, 0);
    }
#if __has_builtin(__builtin_amdgcn_s_wait_asynccnt)
    __builtin_amdgcn_s_wait_asynccnt(0);
#else
    asm volatile("s_wait_asynccnt 0" ::: "memory");
#endif
#else
    for (int i = 0; i < CHUNKS; ++i) {
        const int chunk = i * TPB + tid;
        v4f v = *(const v4f*)(xr + (size_t)chunk * 4);
        *(v4f*)(&row[chunk * 4]) = v;
    }
#endif
    __syncthreads();

    // ---- 4-pass MSB-first radix select for the K-th largest key ---------
    unsigned prefix = 0;
    unsigned want = KSEL;
    for (int pass = 0; pass < 4; ++pass) {
        const int shift = 24 - pass * 8;
        for (int i = tid; i < 256; i += TPB) hist[i] = 0u;
        __syncthreads();

        const unsigned hmask = (pass == 0) ? 0u : (0xFFFFFFFFu << (shift + 8));
        for (int j = tid; j < ROWLEN; j += TPB) {
            const unsigned k = fkey(row[j]);
            if (((k ^ prefix) & hmask) == 0u)
                atomicAdd(&hist[(k >> shift) & 0xFFu], 1u);
        }
        __syncthreads();

        if (tid == 0) {
            unsigned cum = 0;
            for (int b = 255; b >= 0; --b) {
                const unsigned c = hist[b];
                if (cum + c >= want) { s_sel = (unsigned)b; s_want = want - cum; break; }
                cum += c;
            }
        }
        __syncthreads();
        prefix |= s_sel << shift;
        want = s_want;
    }
    const unsigned t_u = prefix;   // exact key of the K-th largest element

    // ---- wave 0: flag the first `want` elements equal to t_u (index order,
    //      matching top_k tie-break) via wave32 ballot ---------------------
    if (tid < 32) {
        unsigned taken = 0;
        for (int base = 0; base < ROWLEN; base += 32) {
            const bool eq = (fkey(row[base + tid]) == t_u);
            const unsigned m = ballot32(eq);
            const unsigned cnt = __popc(m);
            const unsigned need = want - taken;
            unsigned sel;
            if (cnt <= need) {
                sel = m;
            } else {
                const unsigned pos = __popc(m & ((1u << tid) - 1u));
                sel = ballot32(eq && (pos < need));
            }
            if (tid == 0 && sel) flags[base >> 5] = sel;
            taken += cnt;
            if (taken >= want) break;
        }
    }
    __syncthreads();

    // ---- write pass: scatter-as-mask, non-temporal float4 stores --------
    for (int i = 0; i < CHUNKS; ++i) {
        const int chunk = i * TPB + tid;
        const int j = chunk * 4;
        const v4f v = *(const v4f*)(&row[j]);
        const unsigned fw = flags[j >> 5];
        const unsigned bitpos = (unsigned)(j & 31);
        v4f o;
        o.x = ((fkey(v.x) > t_u) || ((fw >> (bitpos + 0)) & 1u)) ? fmaxf(v.x, 0.f) : 0.f;
        o.y = ((fkey(v.y) > t_u) || ((fw >> (bitpos + 1)) & 1u)) ? fmaxf(v.y, 0.f) : 0.f;
        o.z = ((fkey(v.z) > t_u) || ((fw >> (bitpos + 2)) & 1u)) ? fmaxf(v.z, 0.f) : 0.f;
        o.w = ((fkey(v.w) > t_u) || ((fw >> (bitpos + 3)) & 1u)) ? fmaxf(v.w, 0.f) : 0.f;
        __builtin_nontemporal_store(o, (v4f*)(orow + j));
    }
}

extern "C" void kernel_launch(void* const* d_in, const int* in_sizes, int n_in,
                              void* d_out, int out_size, void* d_ws, size_t ws_size,
                              hipStream_t stream) {
    const float* x = (const float*)d_in[0];
    float* out = (float*)d_out;
    const int nrows = in_sizes[0] / ROWLEN;   // 8192
    hipLaunchKernelGGL(topk_scatter_kernel, dim3(nrows), dim3(TPB), 0, stream, x, out);
}